// FocalLoss_74148315398751
// MI455X (gfx1250) — compile-verified
//
#include <hip/hip_runtime.h>
#include <hip/hip_bf16.h>
#include <cstdint>

#define NB 8
#define NA 120000
#define NK 80
#define NM 32
#define EPSV 1e-4f

typedef float v2f __attribute__((ext_vector_type(2)));
typedef float v8f __attribute__((ext_vector_type(8)));
typedef unsigned int u32x4 __attribute__((ext_vector_type(4)));
typedef int i32x4 __attribute__((ext_vector_type(4)));
typedef int i32x8 __attribute__((ext_vector_type(8)));

__device__ __forceinline__ float wave_sum_shfl(float v) {
  for (int off = 16; off; off >>= 1) v += __shfl_xor(v, off, 32);
  return v;
}

// Wave-level sum on the matrix pipe: A per-lane = {s, 0}, B = all ones.
// With the documented 16x4 f32 A layout (lane m: K=0/1, lane m+16: K=2/3),
// D[m,n] = s[m] + s[m+16] regardless of B's layout (B is all ones).
// Summing the 8 D VGPRs gives the two half-wave sums; one xor-16 finishes.
__device__ __forceinline__ float wave_sum_wmma(float v) {
#if __has_builtin(__builtin_amdgcn_wmma_f32_16x16x4_f32)
  v2f a; a.x = v; a.y = 0.0f;
  v2f bm; bm.x = 1.0f; bm.y = 1.0f;
  v8f c = {};
  v8f d = __builtin_amdgcn_wmma_f32_16x16x4_f32(
      false, a, false, bm, (short)0, c, false, false);
  float s = d[0] + d[1] + d[2] + d[3] + d[4] + d[5] + d[6] + d[7];
  s += __shfl_xor(s, 16, 32);
  return s;
#else
  return wave_sum_shfl(v);
#endif
}

// Branch-free focal element: one v_log_f32 per element, selects via cndmask.
__device__ __forceinline__ float focal_elem(float x, bool tpos) {
  const float c = fminf(fmaxf(x, EPSV), 1.0f - EPSV);
  const float u = 1.0f - c;
  const float w  = tpos ? 0.25f * u * u : 0.75f * c * c;
  const float la = tpos ? c : u;
  return w * (-logf(la));
}

// ---------------------------------------------------------------------------
// Kernel 1: per-anchor GT assignment + regression loss partials.
// grid = (ceil(NA/256), NB), block = 256
// ---------------------------------------------------------------------------
__global__ void __launch_bounds__(256) assign_kernel(
    const float* __restrict__ anchors,       // [NA*4]
    const float* __restrict__ regressions,   // [NB*NA*4]
    const float* __restrict__ annotations,   // [NB*NM*5]
    int*   __restrict__ meta,                // [NB*NA]
    float* __restrict__ reg_partial,         // [NB*512]
    float* __restrict__ npos_partial)        // [NB*512]
{
  const int b = blockIdx.y;
  const int tid = threadIdx.x;
  const int a = blockIdx.x * 256 + tid;
  const int ac = (a < NA) ? a : (NA - 1);

  __shared__ float sann[NM * 5];
  __shared__ float sanc[256 * 4];
  __shared__ float sredf[8];
  __shared__ float sredn[8];

  // ---- Stage annotations [32x5] for batch b into LDS via the TDM ----------
#if __has_builtin(__builtin_amdgcn_tensor_load_to_lds)
  if (tid < 32) {
    unsigned long long g =
        (unsigned long long)(const void*)(annotations + (size_t)b * (NM * 5));
    unsigned int lds0 = (unsigned int)(uintptr_t)(void*)&sann[0];
    // D# group0: count=1 | lds_addr | global_addr[56:0] | type=2
    u32x4 g0 = { 1u,
                 lds0,
                 (unsigned)(g & 0xffffffffu),
                 (unsigned)((g >> 32) & 0x01ffffffu) | 0x80000000u };
    // D# group1: data_size=4B, tensor_dim0=160, tensor_dim1=1,
    //            tile_dim0=160, tile_dim1=1, stride0=160
    i32x8 g1 = { (int)(2u << 16),
                 (int)((unsigned)(NM * 5) << 16),
                 (int)(1u << 16),
                 (int)((unsigned)(NM * 5) << 16),
                 1,
                 (int)(NM * 5),
                 0, 0 };
    i32x4 z4 = { 0, 0, 0, 0 };
#if defined(__clang_major__) && (__clang_major__ >= 23)
    i32x8 z8 = { 0, 0, 0, 0, 0, 0, 0, 0 };
    __builtin_amdgcn_tensor_load_to_lds(g0, g1, z4, z4, z8, 0);
#else
    __builtin_amdgcn_tensor_load_to_lds(g0, g1, z4, z4, 0);
#endif
    __builtin_amdgcn_s_wait_tensorcnt(0);
  }
#else
  for (int i = tid; i < NM * 5; i += 256)
    sann[i] = annotations[(size_t)b * (NM * 5) + i];
#endif

  // ---- Stage this lane's anchor box via async global->LDS (ASYNCcnt) ------
  {
    unsigned long long ga =
        (unsigned long long)(const void*)(anchors + (size_t)ac * 4);
    unsigned int ldst = (unsigned int)(uintptr_t)(void*)&sanc[tid * 4];
    asm volatile("global_load_async_to_lds_b128 %0, %1, off"
                 :: "v"(ldst), "v"(ga) : "memory");
    asm volatile("s_wait_asynccnt 0" ::: "memory");
  }
  __syncthreads();

  const float ax0 = sanc[tid * 4 + 0];
  const float ay0 = sanc[tid * 4 + 1];
  const float ax1 = sanc[tid * 4 + 2];
  const float ay1 = sanc[tid * 4 + 3];
  const float aw = ax1 - ax0, ah = ay1 - ay0;
  const float acx = ax0 + 0.5f * aw, acy = ay0 + 0.5f * ah;
  const float aarea = aw * ah;

  float best = -3.0e38f;
  int bestm = 0;
#pragma unroll 4
  for (int m = 0; m < NM; ++m) {
    const float gx0 = sann[m * 5 + 0], gy0 = sann[m * 5 + 1];
    const float gx1 = sann[m * 5 + 2], gy1 = sann[m * 5 + 3];
    const float lbl = sann[m * 5 + 4];
    float iw = fminf(ax1, gx1) - fmaxf(ax0, gx0);
    float ih = fminf(ay1, gy1) - fmaxf(ay0, gy0);
    iw = fmaxf(iw, 0.0f); ih = fmaxf(ih, 0.0f);
    const float inter = iw * ih;
    const float ua = fmaxf(aarea + (gx1 - gx0) * (gy1 - gy0) - inter, 1e-8f);
    float v = inter / ua;
    if (lbl == -1.0f) v = -1.0f;             // padding box
    if (v > best) { best = v; bestm = m; }   // first-max tie-break
  }

  const bool inrange = (a < NA);
  const bool pos = inrange && (best >= 0.5f);
  const bool neg = (best < 0.4f);
  const float* g = &sann[bestm * 5];
  const int lab = (int)g[4];
  if (inrange)
    meta[(size_t)b * NA + a] = (pos ? 2 : (neg ? 1 : 0)) | (lab << 2);

  float rsum = 0.0f;
  if (pos) {
    const float* r = regressions + ((size_t)b * NA + a) * 4;
    const float gw0 = g[2] - g[0], gh0 = g[3] - g[1];
    const float gcx = g[0] + 0.5f * gw0, gcy = g[1] + 0.5f * gh0;
    const float gw = fmaxf(gw0, 1.0f), gh = fmaxf(gh0, 1.0f);
    float rt[4];
    rt[0] = ((gcx - acx) / aw) * 10.0f;
    rt[1] = ((gcy - acy) / ah) * 10.0f;
    rt[2] = logf(gw / aw) * 5.0f;
    rt[3] = logf(gh / ah) * 5.0f;
#pragma unroll
    for (int j = 0; j < 4; ++j) {
      const float d = fabsf(rt[j] - r[j]);
      rsum += (d <= (1.0f / 9.0f)) ? 4.5f * d * d : (d - 0.5f / 9.0f);
    }
  }
  const float npos = pos ? 1.0f : 0.0f;

  const int lane = tid & 31, wid = tid >> 5;
  const float wr = wave_sum_shfl(rsum);
  const float wn = wave_sum_shfl(npos);
  if (lane == 0) { sredf[wid] = wr; sredn[wid] = wn; }
  __syncthreads();
  if (tid == 0) {
    float rr = 0.0f, nn = 0.0f;
    for (int i = 0; i < 8; ++i) { rr += sredf[i]; nn += sredn[i]; }
    reg_partial [b * 512 + blockIdx.x] = rr;
    npos_partial[b * 512 + blockIdx.x] = nn;
  }
}

// ---------------------------------------------------------------------------
// Kernel 2: streaming focal classification loss (307 MB, bandwidth bound).
// grid = (512, NB), block = 256. Grid-stride over float4 quads, manually
// unrolled x2 so each wave keeps two independent 128-bit loads in flight.
// ---------------------------------------------------------------------------
__device__ __forceinline__ float quad_loss(const float4 c4, int m, int k0) {
  const int st = m & 3;                 // 0=ignore 1=neg 2=pos
  if (st == 0) return 0.0f;             // skips the 4 logs entirely
  const bool p = (st == 2);
  const int lab = m >> 2;
  float s;
  s  = focal_elem(c4.x, p && (k0     == lab));
  s += focal_elem(c4.y, p && (k0 + 1 == lab));
  s += focal_elem(c4.z, p && (k0 + 2 == lab));
  s += focal_elem(c4.w, p && (k0 + 3 == lab));
  return s;
}

__global__ void __launch_bounds__(256) focal_cls_kernel(
    const float* __restrict__ cls,         // [NB*NA*NK]
    const int*   __restrict__ meta,        // [NB*NA]
    float*       __restrict__ cls_partial) // [NB*512]
{
  const int b = blockIdx.y;
  const int tid = threadIdx.x;
  const float4* __restrict__ cp = (const float4*)(cls + (size_t)b * NA * NK);
  const int* __restrict__ mb = meta + (size_t)b * NA;

  const int NQ = NA * NK / 4;              // 2,400,000 quads per batch item
  const int stride = gridDim.x * 256;
  float acc = 0.0f;

  int q = blockIdx.x * 256 + tid;
  for (; q + stride < NQ; q += 2 * stride) {
    const int q1 = q + stride;
    const int a0 = q  / 20, a1 = q1 / 20;  // 20 quads per anchor (80/4)
    // issue both 128-bit loads + both meta loads before any math
    const float4 c0 = cp[q];
    const float4 c1 = cp[q1];
    const int m0 = mb[a0];
    const int m1 = mb[a1];
    acc += quad_loss(c0, m0, (q  - a0 * 20) * 4);
    acc += quad_loss(c1, m1, (q1 - a1 * 20) * 4);
  }
  if (q < NQ) {
    const int a0 = q / 20;
    acc += quad_loss(cp[q], mb[a0], (q - a0 * 20) * 4);
  }

  // wave reduction on the matrix pipe (EXEC is all-ones here: reconverged)
  const float w = wave_sum_wmma(acc);

  __shared__ float sred[8];
  const int lane = tid & 31, wid = tid >> 5;
  if (lane == 0) sred[wid] = w;
  __syncthreads();
  if (tid == 0) {
    float s = 0.0f;
    for (int i = 0; i < 8; ++i) s += sred[i];
    cls_partial[b * 512 + blockIdx.x] = s;
  }
}

// ---------------------------------------------------------------------------
// Kernel 3: final deterministic reduction -> 2 output scalars.
// grid = 1, block = 256 (wave w handles batch item w).
// ---------------------------------------------------------------------------
__global__ void __launch_bounds__(256) finalize_kernel(
    const float* __restrict__ cls_partial,
    const float* __restrict__ reg_partial,
    const float* __restrict__ npos_partial,
    int nblk1,
    float* __restrict__ out)
{
  const int tid = threadIdx.x, lane = tid & 31, wid = tid >> 5;
  __shared__ float cl[8], rl[8];

  float cs = 0.0f, rs = 0.0f, ns = 0.0f;
  for (int i = lane; i < 512; i += 32) cs += cls_partial[wid * 512 + i];
  for (int i = lane; i < nblk1; i += 32) {
    rs += reg_partial [wid * 512 + i];
    ns += npos_partial[wid * 512 + i];
  }
  for (int off = 16; off; off >>= 1) {
    cs += __shfl_xor(cs, off, 32);
    rs += __shfl_xor(rs, off, 32);
    ns += __shfl_xor(ns, off, 32);
  }
  if (lane == 0) {
    cl[wid] = cs / fmaxf(ns, 1.0f);
    rl[wid] = (ns > 0.0f) ? (rs / fmaxf(ns * 4.0f, 1.0f)) : 0.0f;
  }
  __syncthreads();
  if (tid == 0) {
    float c = 0.0f, r = 0.0f;
    for (int i = 0; i < 8; ++i) { c += cl[i]; r += rl[i]; }
    out[0] = c * 0.125f;
    out[1] = r * 0.125f;
  }
}

// ---------------------------------------------------------------------------
extern "C" void kernel_launch(void* const* d_in, const int* in_sizes, int n_in,
                              void* d_out, int out_size, void* d_ws, size_t ws_size,
                              hipStream_t stream) {
  (void)in_sizes; (void)n_in; (void)out_size; (void)ws_size;
  const float* classifications = (const float*)d_in[0];  // [8,120000,80]
  const float* regressions     = (const float*)d_in[1];  // [8,120000,4]
  const float* anchors         = (const float*)d_in[2];  // [1,120000,4]
  const float* annotations     = (const float*)d_in[3];  // [8,32,5]
  float* out = (float*)d_out;

  char* ws = (char*)d_ws;
  float* cls_partial  = (float*)(ws);            // 8*512 f32
  float* reg_partial  = (float*)(ws + 16384);    // 8*512 f32
  float* npos_partial = (float*)(ws + 32768);    // 8*512 f32
  int*   meta         = (int*)  (ws + 49152);    // 8*120000 i32 (~3.84 MB)

  const int nblk1 = (NA + 255) / 256;            // 469

  assign_kernel<<<dim3(nblk1, NB), 256, 0, stream>>>(
      anchors, regressions, annotations, meta, reg_partial, npos_partial);
  focal_cls_kernel<<<dim3(512, NB), 256, 0, stream>>>(
      classifications, meta, cls_partial);
  finalize_kernel<<<1, 256, 0, stream>>>(
      cls_partial, reg_partial, npos_partial, nblk1, out);
}